// kernel_generated_0_21887153341304
// MI455X (gfx1250) — compile-verified
//
#include <hip/hip_runtime.h>

typedef __attribute__((ext_vector_type(2))) float v2f;
typedef __attribute__((ext_vector_type(4))) float v4f;
typedef __attribute__((ext_vector_type(8))) float v8f;
typedef __attribute__((ext_vector_type(3), aligned(4))) float v3f;

namespace {
constexpr int HW = 56;
constexpr int SP = HW * HW;            // 3136 spatial positions
constexpr int LCOUNT = 32;             // l dimension
constexpr int TILES_PER_L = SP / 16;   // 196
}

__global__ __launch_bounds__(256)
void fused_shift_unfold_gemm(const float* __restrict__ x,
                             const float* __restrict__ wgt,
                             float* __restrict__ out) {
    const int lane = (int)(threadIdx.x & 31u);
    const int wave = (int)(threadIdx.x >> 5u);
    // gtile is wave-uniform: scalarize so l/tile math is SALU.
    const int gt   = __builtin_amdgcn_readfirstlane((int)blockIdx.x * 8 + wave);
    const int l    = gt / TILES_PER_L;
    const int tile = gt - l * TILES_PER_L;

    const bool h    = (lane >> 4) != 0;          // K sub-pair select
    const int  mrow = lane & 15;                 // A row (M) / B,D column (N)
    const int  s    = tile * 16 + mrow;
    const int  p    = s / HW;
    const int  o    = s - p * HW;
    const int  rowbase = ((p + HW - 1) % HW) * HW;   // output roll(+1) over height

    // Width taps: q(i) = (o+i-2) mod 56 (unfold + roll(+1)); all in [0,56) -> safe loads.
    int q0 = o - 2; q0 += (q0 < 0) ? HW : 0;     // i=0
    int q1 = o - 1; q1 += (q1 < 0) ? HW : 0;     // i=1
    const int q2 = o;                            // i=2
    const bool vo0 = o >= 1;                     // i=0 unfold gate
    const bool vo2 = o <= HW - 2;                // i=2 unfold gate
    // Per (i_half0|i_half1) pattern, one combined lane-mask:
    const bool g02 = h ? vo2 : vo0;
    const bool g10 = h ? vo0 : true;
    const bool g21 = h ? true : vo2;

    // Clamped channel bases (element offsets). c2 = l+k-1; OOB zeroing done on B side.
    int base[3][2];
    #pragma unroll
    for (int k = 0; k < 3; ++k) {
        const int c2c = min(max(l + k - 1, 0), LCOUNT - 1);
        base[k][0] = (c2c * 2 + 0) * SP + rowbase;
        base[k][1] = (c2c * 2 + 1) * SP + rowbase;
    }

    // Weight triples: w[k,j,l,m,i] -> float offset (k*2+j)*384 + l*12 + m*3 + i.
    // One b96 per (k,j); columns >= 4 are don't-care (never stored) -> mrow&3 keeps it in-bounds.
    const float* wl = wgt + l * 12;
    const int nb = (mrow & 3) * 3;
    v3f W[3][2];
    #pragma unroll
    for (int k = 0; k < 3; ++k)
        #pragma unroll
        for (int j = 0; j < 2; ++j)
            W[k][j] = *(const v3f*)(wl + (k * 2 + j) * (LCOUNT * 12) + nb);
    // Uniform l-boundary gate applied once to whole k-groups (replaces 20 per-lane masks):
    const v3f z3 = {0.0f, 0.0f, 0.0f};
    if (l == 0)          { W[0][0] = z3; W[0][1] = z3; }
    if (l == LCOUNT - 1) { W[2][0] = z3; W[2][1] = z3; }

    // ---- Issue ALL x-loads unconditionally (addresses always in-range); gate later. ----
    const float* xhi = x + 64 * SP;              // folded channel pair: x[c] + x[c+64]
    auto aelem = [&](int bA, int bB, int qA, int qB) -> float {
        const int idx = (h ? bB : bA) + (h ? qB : qA);
        const float lo = x[idx];                 // unconditional loads -> clause + pipeline
        const float hi = xhi[idx];
        return lo + hi;
    };
    const float v00 = aelem(base[0][0], base[0][0], q0, q2);   // t {0 |2 }
    const float v01 = aelem(base[0][0], base[0][1], q1, q0);   // t {1 |3 }
    const float v10 = aelem(base[0][1], base[1][0], q1, q0);   // t {4 |6 }
    const float v11 = aelem(base[0][1], base[1][0], q2, q1);   // t {5 |7 }
    const float v20 = aelem(base[1][0], base[1][1], q2, q1);   // t {8 |10}
    const float v21 = aelem(base[1][1], base[1][1], q0, q2);   // t {9 |11}
    const float v30 = aelem(base[2][0], base[2][0], q0, q2);   // t {12|14}
    const float v31 = aelem(base[2][0], base[2][1], q1, q0);   // t {13|15}
    const float v40 = aelem(base[2][1], base[2][1], q1, q1);   // t {16|18d}
    const float v41 = aelem(base[2][1], base[2][1], q2, q2);   // t {17|19d}

    v8f acc = {};
    v2f a, b;

    // kstep 0
    a.x = g02 ? v00 : 0.0f;
    a.y = g10 ? v01 : 0.0f;
    b.x = h ? W[0][0].z : W[0][0].x;
    b.y = h ? W[0][1].x : W[0][0].y;
    acc = __builtin_amdgcn_wmma_f32_16x16x4_f32(false, a, false, b, (short)0, acc, false, false);

    // kstep 1
    a.x = g10 ? v10 : 0.0f;
    a.y = g21 ? v11 : 0.0f;
    b.x = h ? W[1][0].x : W[0][1].y;
    b.y = h ? W[1][0].y : W[0][1].z;
    acc = __builtin_amdgcn_wmma_f32_16x16x4_f32(false, a, false, b, (short)0, acc, false, false);

    // kstep 2
    a.x = g21 ? v20 : 0.0f;
    a.y = g02 ? v21 : 0.0f;
    b.x = h ? W[1][1].y : W[1][0].z;
    b.y = h ? W[1][1].z : W[1][1].x;
    acc = __builtin_amdgcn_wmma_f32_16x16x4_f32(false, a, false, b, (short)0, acc, false, false);

    // kstep 3
    a.x = g02 ? v30 : 0.0f;
    a.y = g10 ? v31 : 0.0f;
    b.x = h ? W[2][0].z : W[2][0].x;
    b.y = h ? W[2][1].x : W[2][0].y;
    acc = __builtin_amdgcn_wmma_f32_16x16x4_f32(false, a, false, b, (short)0, acc, false, false);

    // kstep 4 (half==1 lanes carry dead t=18,19: zeroed on B only; A garbage is harmless)
    a.x = v40;
    a.y = g21 ? v41 : 0.0f;
    b.x = h ? 0.0f : W[2][1].y;
    b.y = h ? 0.0f : W[2][1].z;
    acc = __builtin_amdgcn_wmma_f32_16x16x4_f32(false, a, false, b, (short)0, acc, false, false);

    // D layout: lane holds column N=mrow for rows M = vr + 8*half; only N<4 is real output.
    // 8 consecutive spatial floats, 32B aligned -> two b128 stores, 32-bit offsets.
    if (mrow < 4) {
        const unsigned ooff = (unsigned)((l * 4 + mrow) * SP + tile * 16 + (h ? 8 : 0));
        v4f lo4 = { acc[0], acc[1], acc[2], acc[3] };
        v4f hi4 = { acc[4], acc[5], acc[6], acc[7] };
        *(v4f*)(out + ooff)     = lo4;
        *(v4f*)(out + ooff + 4) = hi4;
    }
}

extern "C" void kernel_launch(void* const* d_in, const int* /*in_sizes*/, int /*n_in*/,
                              void* d_out, int /*out_size*/, void* /*d_ws*/, size_t /*ws_size*/,
                              hipStream_t stream) {
    const float* x = (const float*)d_in[0];   // (1,128,56,56) f32
    const float* w = (const float*)d_in[1];   // (3,2,32,4,3)  f32
    float* out = (float*)d_out;               // (1,128,56,56) f32

    const int total_waves = LCOUNT * TILES_PER_L;   // 6272
    dim3 grid(total_waves / 8);                     // 784 blocks x 8 waves
    dim3 block(256);
    hipLaunchKernelGGL(fused_shift_unfold_gemm, grid, block, 0, stream, x, w, out);
}